// GraphAttentionLayer_67224828117276
// MI455X (gfx1250) — compile-verified
//
#include <hip/hip_runtime.h>
#include <hip/hip_bf16.h>

typedef __attribute__((ext_vector_type(2))) float v2f;
typedef __attribute__((ext_vector_type(8))) float v8f;

#define B_    8
#define N_    2048
#define FIN   256
#define FOUT  256
#define NEGINF   (-1e9f)
#define SLOPE    0.2f
#define RUNMIN   (-3.0e38f)

// ---------------------------------------------------------------------------
// Kernel A: h = x @ W^T   (M = B*N = 16384, K = 256, N = 256), f32 WMMA.
// One wave computes one 16x16 output tile; 64 v_wmma_f32_16x16x4_f32 per tile.
// ---------------------------------------------------------------------------
__global__ __launch_bounds__(128) void gat_h_gemm(const float* __restrict__ x,
                                                  const float* __restrict__ W,
                                                  float* __restrict__ h) {
    const int lane = threadIdx.x & 31;
    const int wave = threadIdx.x >> 5;
    const int tile = blockIdx.x * 4 + wave;      // 16384 tiles total
    const int rowtile = tile >> 4;               // 1024 row tiles
    const int coltile = tile & 15;               // 16 col tiles
    const int m    = lane & 15;                  // A: M row / B: N col / D: col
    const int half = lane >> 4;

    // A operand: A[m][k], k = half*2 + v  -> contiguous float2 in x row
    const float* arow = x + (rowtile * 16 + m) * FIN + half * 2;
    // B operand: B[k][n] = W[n][k], n = lane&15, k = half*2 + v -> float2 in W row
    const float* brow = W + (coltile * 16 + m) * FIN + half * 2;

    v8f acc = {};
    #pragma unroll 8
    for (int k = 0; k < FIN; k += 4) {
        v2f a = *(const v2f*)(arow + k);
        v2f b = *(const v2f*)(brow + k);
        acc = __builtin_amdgcn_wmma_f32_16x16x4_f32(false, a, false, b,
                                                    (short)0, acc, false, false);
    }
    // D layout: row = v + 8*half, col = lane&15
    float* orow = h + (rowtile * 16 + 8 * half) * FOUT + coltile * 16 + m;
    #pragma unroll
    for (int v = 0; v < 8; ++v) orow[v * FOUT] = acc[v];
}

// ---------------------------------------------------------------------------
// Kernel B: s_src = h @ a_src, s_dst = h @ a_dst   (per-node dot products)
// ---------------------------------------------------------------------------
__global__ __launch_bounds__(256) void gat_scores(const float* __restrict__ h,
                                                  const float* __restrict__ a_src,
                                                  const float* __restrict__ a_dst,
                                                  float* __restrict__ s_src,
                                                  float* __restrict__ s_dst) {
    __shared__ float as[FOUT];
    __shared__ float ad[FOUT];
    for (int i = threadIdx.x; i < FOUT; i += 256) { as[i] = a_src[i]; ad[i] = a_dst[i]; }
    __syncthreads();

    const int node = blockIdx.x * 256 + threadIdx.x;   // < B*N
    const float* hr = h + node * FOUT;
    float ss = 0.f, sd = 0.f;
    #pragma unroll 4
    for (int i = 0; i < FOUT; i += 4) {
        float4 hv = *(const float4*)(hr + i);
        ss += hv.x * as[i] + hv.y * as[i + 1] + hv.z * as[i + 2] + hv.w * as[i + 3];
        sd += hv.x * ad[i] + hv.y * ad[i + 1] + hv.z * ad[i + 2] + hv.w * ad[i + 3];
    }
    s_src[node] = ss;
    s_dst[node] = sd;
}

// ---------------------------------------------------------------------------
// Kernel B2: tmax[b] = max_j s_dst[b][j]   (per-batch reduction, 8 blocks)
// Gives an exact unmasked row max M_i = leaky(s_src_i + tmax_b) so the
// softmax needs no online max tracking (leaky is monotone increasing).
// ---------------------------------------------------------------------------
__global__ __launch_bounds__(256) void gat_tmax(const float* __restrict__ s_dst,
                                                float* __restrict__ tmax) {
    __shared__ float red[256];
    const int b = blockIdx.x;
    float m = RUNMIN;
    for (int j = threadIdx.x; j < N_; j += 256) m = fmaxf(m, s_dst[b * N_ + j]);
    red[threadIdx.x] = m;
    __syncthreads();
    #pragma unroll
    for (int s = 128; s > 0; s >>= 1) {
        if (threadIdx.x < s) red[threadIdx.x] = fmaxf(red[threadIdx.x], red[threadIdx.x + s]);
        __syncthreads();
    }
    if (threadIdx.x == 0) tmax[b] = red[0];
}

// ---------------------------------------------------------------------------
// Kernel C: fused  p = exp(mask(leaky(s_i + t_j)) - M_i) -> out = (P @ h)/sum
// Workgroup = (batch b, 16-row i-tile); the 4 waves split the *j* dimension
// (512 j's each, all 256 output features) so score work is done exactly once.
// Fixed softmax shift M_i => partial (acc, sum) merge by plain addition via
// ds_add_f32 LDS atomics at the end.
// ---------------------------------------------------------------------------
__global__ __launch_bounds__(128) void gat_attend(const float* __restrict__ h,
                                                  const int*   __restrict__ adj,
                                                  const float* __restrict__ s_src,
                                                  const float* __restrict__ s_dst,
                                                  const float* __restrict__ tmax,
                                                  float* __restrict__ out) {
    // P staging: padded 16x18 so transposed float2 reads stay 8B-aligned
    __shared__ float Plds[4][16][18];
    __shared__ float mbuf[16][FOUT];          // cross-wave accumulator merge
    __shared__ float sums[16];                // cross-wave row-sum merge

    const int lane = threadIdx.x & 31;
    const int wave = threadIdx.x >> 5;
    const int b     = blockIdx.x >> 7;        // / (N/16)
    const int itile = blockIdx.x & 127;
    const int i0    = itile * 16;
    const int n     = lane & 15;
    const int half  = lane >> 4;

    // zero the merge buffers
    for (int idx = threadIdx.x; idx < 16 * FOUT; idx += 128)
        mbuf[idx >> 8][idx & (FOUT - 1)] = 0.f;
    if (threadIdx.x < 16) sums[threadIdx.x] = 0.f;
    __syncthreads();

    const float* hb   = h + b * N_ * FOUT;
    const int*   adjb = adj + (b * N_ + i0) * N_;
    const float  tb   = tmax[b];

    // per-row softmax shift M_i (D-layout rows m = v + 8*half for this lane)
    float ssrc[8], Mrow[8];
    #pragma unroll
    for (int v = 0; v < 8; ++v) {
        const float s = s_src[b * N_ + i0 + v + 8 * half];
        float em = s + tb;
        em = em > 0.f ? em : SLOPE * em;      // exact unmasked row max
        ssrc[v] = s;
        Mrow[v] = em;
    }

    float rsum[8];
    #pragma unroll
    for (int v = 0; v < 8; ++v) rsum[v] = 0.f;

    v8f acc[16];
    #pragma unroll
    for (int t = 0; t < 16; ++t) acc[t] = (v8f){};

    const int jt0 = wave * (N_ / 16 / 4);     // this wave's 32 j-tiles
    for (int jt = jt0; jt < jt0 + N_ / 16 / 4; ++jt) {
        const int j0 = jt * 16;

        // prefetch adjacency stream (the only HBM-resident input) ahead
        if (jt + 4 < jt0 + N_ / 16 / 4)
            __builtin_prefetch(adjb + (8 * half) * N_ + (jt + 4) * 16 + n, 0, 1);

        // ---- probabilities in C/D layout: row m = v+8*half, col j0+n ------
        const float sd = s_dst[b * N_ + j0 + n];
        #pragma unroll
        for (int v = 0; v < 8; ++v) {
            const int m = v + 8 * half;
            const int a = adjb[m * N_ + j0 + n];
            float e = ssrc[v] + sd;
            e = e > 0.f ? e : SLOPE * e;                  // leaky relu
            const float pv = (a != 0) ? __expf(e - Mrow[v]) : 0.f;
            rsum[v] += pv;
            Plds[wave][m][n] = pv;
        }
        // per-wave private LDS buffer; DS ops are in-order within a wave.

        // ---- acc += P (16x16) x Htile (16x256) via 4 K-chunks of f32 WMMA -
        #pragma unroll
        for (int kk = 0; kk < 4; ++kk) {
            const int krow = kk * 4 + half * 2;
            // A operand: P[m = lane&15][k = krow + v]  (transposed read)
            v2f a = *(const v2f*)&Plds[wave][n][krow];
            const float* h0 = hb + (j0 + krow) * FOUT + n;
            #pragma unroll
            for (int t = 0; t < 16; ++t) {
                v2f bb;
                bb.x = h0[t * 16];            // B[k=krow  ][col]
                bb.y = h0[t * 16 + FOUT];     // B[k=krow+1][col]
                acc[t] = __builtin_amdgcn_wmma_f32_16x16x4_f32(
                    false, a, false, bb, (short)0, acc[t], false, false);
            }
        }
    }

    // ---- merge partials across waves (plain adds: shared fixed shift M) ---
    #pragma unroll
    for (int v = 0; v < 8; ++v)
        atomicAdd(&sums[v + 8 * half], rsum[v]);          // ds_add_f32
    #pragma unroll
    for (int t = 0; t < 16; ++t)
        #pragma unroll
        for (int v = 0; v < 8; ++v)
            atomicAdd(&mbuf[v + 8 * half][t * 16 + n], acc[t][v]);
    __syncthreads();

    // ---- normalize and store (coalesced) ----------------------------------
    for (int idx = threadIdx.x; idx < 16 * FOUT; idx += 128) {
        const int r = idx >> 8;               // FOUT == 256
        const int c = idx & (FOUT - 1);
        out[(b * N_ + i0 + r) * FOUT + c] = mbuf[r][c] / sums[r];
    }
}

// ---------------------------------------------------------------------------
extern "C" void kernel_launch(void* const* d_in, const int* in_sizes, int n_in,
                              void* d_out, int out_size, void* d_ws, size_t ws_size,
                              hipStream_t stream) {
    const float* x     = (const float*)d_in[0];
    const int*   adj   = (const int*)  d_in[1];
    const float* W     = (const float*)d_in[2];
    const float* a_src = (const float*)d_in[3];
    const float* a_dst = (const float*)d_in[4];
    float* out = (float*)d_out;

    // workspace layout: h (B*N*FOUT f32), s_src (B*N), s_dst (B*N), tmax (B)
    float* h     = (float*)d_ws;
    float* s_src = h + (size_t)B_ * N_ * FOUT;
    float* s_dst = s_src + (size_t)B_ * N_;
    float* tmaxp = s_dst + (size_t)B_ * N_;

    // 1) h = x @ W^T : 16384 16x16 tiles, 4 waves per block
    gat_h_gemm<<<(B_ * N_ / 16) * (FOUT / 16) / 4, 128, 0, stream>>>(x, W, h);
    // 2) per-node attention score contributions
    gat_scores<<<B_ * N_ / 256, 256, 0, stream>>>(h, a_src, a_dst, s_src, s_dst);
    // 2b) per-batch max of s_dst -> exact softmax shift
    gat_tmax<<<B_, 256, 0, stream>>>(s_dst, tmaxp);
    // 3) fused masked-softmax attention + aggregation (j split across waves)
    gat_attend<<<B_ * (N_ / 16), 128, 0, stream>>>(h, adj, s_src, s_dst, tmaxp, out);
}